// GNNModel_68341519614046
// MI455X (gfx1250) — compile-verified
//
#include <hip/hip_runtime.h>
#include <hip/hip_bf16.h>

typedef __attribute__((ext_vector_type(2))) float v2f;
typedef __attribute__((ext_vector_type(8))) float v8f;

__device__ __forceinline__ float swish_f(float z) {
    return z / (1.0f + __expf(-z));
}

// ---------------------------------------------------------------------------
// C[M,64] = A[M,64] @ W[64,64]   (raw product; GCN adds bias after aggregation)
// One wave -> 16x64 tile via 4 accumulators of V_WMMA_F32_16X16X4_F32.
// W staged in LDS in K-pair-interleaved layout so each lane's B fragment
// {W[k][n], W[k+1][n]} is one aligned ds_load_b64 (no VGPR shuffling).
//   sW[(k>>1)*128 + (n<<1) + (k&1)] = W[k][n]
// ---------------------------------------------------------------------------
__global__ __launch_bounds__(256) void gemm_n64_f32wmma(
    const float* __restrict__ A, const float* __restrict__ W,
    float* __restrict__ C, int M) {
    __shared__ float sW[64 * 64];
    for (int i = threadIdx.x; i < 64 * 64; i += 256) {
        const int k = i >> 6, n = i & 63;
        sW[(k >> 1) * 128 + (n << 1) + (k & 1)] = W[i];
    }
    __syncthreads();

    const int tile = blockIdx.x * 8 + (threadIdx.x >> 5);
    if (tile * 16 >= M) return;                  // wave-uniform exit: EXEC all-1 for WMMA

    const int lane = threadIdx.x & 31;
    const int hf   = lane >> 4;                  // 0: lanes 0-15, 1: lanes 16-31
    const int mr   = lane & 15;

    const float* arow = A + (size_t)(tile * 16 + mr) * 64;

    v8f c0 = {}, c1 = {}, c2 = {}, c3 = {};
#pragma unroll
    for (int kk = 0; kk < 64; kk += 4) {
        const int kp = (kk >> 1) + hf;           // K-pair index handled by this half-wave
        v2f a = *(const v2f*)(arow + 2 * kp);    // A: M=mr, K={2kp, 2kp+1}
        const float* wrow = sW + kp * 128 + 2 * mr;
        v2f b0 = *(const v2f*)(wrow +  0);
        v2f b1 = *(const v2f*)(wrow + 32);
        v2f b2 = *(const v2f*)(wrow + 64);
        v2f b3 = *(const v2f*)(wrow + 96);
        c0 = __builtin_amdgcn_wmma_f32_16x16x4_f32(false, a, false, b0, (short)0, c0, false, false);
        c1 = __builtin_amdgcn_wmma_f32_16x16x4_f32(false, a, false, b1, (short)0, c1, false, false);
        c2 = __builtin_amdgcn_wmma_f32_16x16x4_f32(false, a, false, b2, (short)0, c2, false, false);
        c3 = __builtin_amdgcn_wmma_f32_16x16x4_f32(false, a, false, b3, (short)0, c3, false, false);
    }

    float* crow = C + (size_t)tile * 16 * 64;
#pragma unroll
    for (int r = 0; r < 8; ++r) {
        const int orow = r + 8 * hf;             // D layout: VGPR r -> M=r (lo half) / r+8 (hi half)
        crow[orow * 64 +  0 + mr] = c0[r];
        crow[orow * 64 + 16 + mr] = c1[r];
        crow[orow * 64 + 32 + mr] = c2[r];
        crow[orow * 64 + 48 + mr] = c3[r];
    }
}

// ---------------------------------------------------------------------------
// D[M,100] = swish(A[M,64] @ Wd[64,100] + bd)  -- N padded to 112 (7 tiles)
// Same K-pair-interleaved LDS layout, stride 224 floats per K-pair row.
// ---------------------------------------------------------------------------
__global__ __launch_bounds__(256) void gemm_dense_swish(
    const float* __restrict__ A, const float* __restrict__ Wd,
    const float* __restrict__ bd, float* __restrict__ D, int M) {
    __shared__ float sW[32 * 224];               // 28 KB of 320 KB WGP LDS
    __shared__ float sb[112];
    for (int i = threadIdx.x; i < 64 * 112; i += 256) {
        const int k = i / 112, n = i % 112;
        sW[(k >> 1) * 224 + (n << 1) + (k & 1)] = (n < 100) ? Wd[k * 100 + n] : 0.0f;
    }
    if (threadIdx.x < 112) sb[threadIdx.x] = (threadIdx.x < 100) ? bd[threadIdx.x] : 0.0f;
    __syncthreads();

    const int tile = blockIdx.x * 8 + (threadIdx.x >> 5);
    if (tile * 16 >= M) return;

    const int lane = threadIdx.x & 31;
    const int hf   = lane >> 4;
    const int mr   = lane & 15;

    const float* arow = A + (size_t)(tile * 16 + mr) * 64;

    v8f zero = {};
    v8f acc[7];
#pragma unroll
    for (int t = 0; t < 7; ++t) acc[t] = zero;

#pragma unroll
    for (int kk = 0; kk < 64; kk += 4) {
        const int kp = (kk >> 1) + hf;
        v2f a = *(const v2f*)(arow + 2 * kp);
        const float* wrow = sW + kp * 224 + 2 * mr;
#pragma unroll
        for (int t = 0; t < 7; ++t) {
            v2f b = *(const v2f*)(wrow + t * 32);
            acc[t] = __builtin_amdgcn_wmma_f32_16x16x4_f32(false, a, false, b, (short)0, acc[t], false, false);
        }
    }

#pragma unroll
    for (int t = 0; t < 7; ++t) {
        const int col = t * 16 + mr;
        if (col < 100) {
            const float bias = sb[col];
#pragma unroll
            for (int r = 0; r < 8; ++r) {
                const int orow = tile * 16 + r + 8 * hf;
                D[(size_t)orow * 100 + col] = swish_f(acc[t][r] + bias);
            }
        }
    }
}

// ---------------------------------------------------------------------------
// Edge scatter: AGG[dst] += H[src] * w   (16 threads/edge, float4 gather,
// 4x global_atomic_add_f32 resolved in the 192MB L2)
// ---------------------------------------------------------------------------
__global__ __launch_bounds__(256) void scatter_edges(
    const float* __restrict__ H, const int* __restrict__ esrc,
    const int* __restrict__ edst, const float* __restrict__ ew,
    float* __restrict__ AGG, int E) {
    const int gid = blockIdx.x * 256 + threadIdx.x;
    const int e = gid >> 4;
    if (e >= E) return;
    const int f = (gid & 15) * 4;
    const int s = esrc[e];
    const int d = edst[e];
    const float w = ew[e];
    const float4 v = *(const float4*)(H + (size_t)s * 64 + f);
    float* out = AGG + (size_t)d * 64 + f;
    atomicAdd(out + 0, v.x * w);
    atomicAdd(out + 1, v.y * w);
    atomicAdd(out + 2, v.z * w);
    atomicAdd(out + 3, v.w * w);
}

__global__ __launch_bounds__(256) void fill_zero(float* __restrict__ p, int n) {
    const int gid = blockIdx.x * 256 + threadIdx.x;
    if (gid < n) p[gid] = 0.0f;
}

// out[i] = swish(in[i] + b[i % 64])
__global__ __launch_bounds__(256) void bias_swish64(
    const float* __restrict__ in, const float* __restrict__ b,
    float* __restrict__ out, int n) {
    const int gid = blockIdx.x * 256 + threadIdx.x;
    if (gid < n) {
        const float z = in[gid] + b[gid & 63];
        out[gid] = swish_f(z);
    }
}

// ---------------------------------------------------------------------------
// out[node] = sigmoid(D[node,:100] . Wo + bo)  -- one wave32 per node,
// coalesced lane loads + shuffle reduction.
// ---------------------------------------------------------------------------
__global__ __launch_bounds__(256) void out_layer(
    const float* __restrict__ D, const float* __restrict__ Wo,
    const float* __restrict__ bo, float* __restrict__ out, int M) {
    const int node = blockIdx.x * 8 + (threadIdx.x >> 5);
    if (node >= M) return;
    const int lane = threadIdx.x & 31;
    const float* row = D + (size_t)node * 100;
    float sum = row[lane]      * Wo[lane]
              + row[lane + 32] * Wo[lane + 32]
              + row[lane + 64] * Wo[lane + 64];
    if (lane < 4) sum += row[96 + lane] * Wo[96 + lane];
#pragma unroll
    for (int off = 16; off > 0; off >>= 1) sum += __shfl_down(sum, off, 32);
    if (lane == 0) {
        const float z = sum + bo[0];
        out[node] = 1.0f / (1.0f + __expf(-z));
    }
}

// ---------------------------------------------------------------------------
extern "C" void kernel_launch(void* const* d_in, const int* in_sizes, int n_in,
                              void* d_out, int out_size, void* d_ws, size_t ws_size,
                              hipStream_t stream) {
    const float* x    = (const float*)d_in[0];
    const int*   esrc = (const int*)  d_in[1];
    const int*   edst = (const int*)  d_in[2];
    const float* ew   = (const float*)d_in[3];
    const float* W1   = (const float*)d_in[4];
    const float* b1   = (const float*)d_in[5];
    const float* W2   = (const float*)d_in[6];
    const float* b2   = (const float*)d_in[7];
    const float* Wd   = (const float*)d_in[8];
    const float* bd   = (const float*)d_in[9];
    const float* Wo   = (const float*)d_in[10];
    const float* bo   = (const float*)d_in[11];
    float* out = (float*)d_out;

    const int M = in_sizes[0] / 64;   // 50000
    const int E = in_sizes[1];        // 800000
    const int nfeat = M * 64;

    // scratch layout: bufA[M*64] | bufB[M*64] | bufC[M*100]  (~45.6 MB)
    float* bufA = (float*)d_ws;
    float* bufB = bufA + (size_t)M * 64;
    float* bufC = bufB + (size_t)M * 64;

    const int tiles       = (M + 15) / 16;
    const int gemm_blocks = (tiles + 7) / 8;
    const int fill_blocks = (nfeat + 255) / 256;
    const int scat_blocks = (int)(((size_t)E * 16 + 255) / 256);
    const int node_blocks = (M + 7) / 8;

    // --- GCN layer 1 ---
    gemm_n64_f32wmma<<<gemm_blocks, 256, 0, stream>>>(x, W1, bufA, M);
    fill_zero<<<fill_blocks, 256, 0, stream>>>(bufB, nfeat);
    scatter_edges<<<scat_blocks, 256, 0, stream>>>(bufA, esrc, edst, ew, bufB, E);
    bias_swish64<<<fill_blocks, 256, 0, stream>>>(bufB, b1, bufA, nfeat);

    // --- GCN layer 2 ---
    gemm_n64_f32wmma<<<gemm_blocks, 256, 0, stream>>>(bufA, W2, bufB, M);
    fill_zero<<<fill_blocks, 256, 0, stream>>>(bufA, nfeat);
    scatter_edges<<<scat_blocks, 256, 0, stream>>>(bufB, esrc, edst, ew, bufA, E);
    bias_swish64<<<fill_blocks, 256, 0, stream>>>(bufA, b2, bufB, nfeat);

    // --- Dense(100, swish) + Dense(1, sigmoid) ---
    gemm_dense_swish<<<gemm_blocks, 256, 0, stream>>>(bufB, Wd, bd, bufC, M);
    out_layer<<<node_blocks, 256, 0, stream>>>(bufC, Wo, bo, out, M);
}